// FocalLoss_3470333575643
// MI455X (gfx1250) — compile-verified
//
#include <hip/hip_runtime.h>
#include <hip/hip_bf16.h>
#include <math.h>

typedef float v2f __attribute__((ext_vector_type(2)));
typedef float v8f __attribute__((ext_vector_type(8)));

#define BG_CLASS   80.0f
#define F_ALPHA    0.25f

// ---------------------------------------------------------------------------
// Kernel 0: zero the accumulator region of the workspace (deterministic init
// every call; harness does not re-poison between graph replays).
// ---------------------------------------------------------------------------
__global__ void fl_init_kernel(float* __restrict__ acc, int n) {
    int i = blockIdx.x * blockDim.x + threadIdx.x;
    if (i < n) acc[i] = 0.0f;
}

// ---------------------------------------------------------------------------
// Kernel 1: per-anchor IoU assignment, class targets, smooth-L1 reg loss.
// Grid: (ceil(A/256), B). Positives are extremely rare (exact-tile matches),
// so per-thread predicated atomics are effectively free.
// ---------------------------------------------------------------------------
__global__ void fl_assign_reg_kernel(const float* __restrict__ regs,     // (B,A,4)
                                     const float* __restrict__ anchors,  // (A,4)
                                     const float* __restrict__ ann,      // (B,G,5)
                                     float* __restrict__ targets,        // (B*A)
                                     float* __restrict__ regsum,         // (B)
                                     float* __restrict__ nposv,          // (B)
                                     int A, int G) {
    __shared__ float sann[16 * 5];
    int b = blockIdx.y;
    for (int i = threadIdx.x; i < G * 5; i += blockDim.x)
        sann[i] = ann[(long)b * G * 5 + i];
    __syncthreads();

    int a = blockIdx.x * blockDim.x + threadIdx.x;
    if (a >= A) return;

    float4 an = *(const float4*)(anchors + 4L * a);   // x0,y0,x1,y1
    float aw  = an.z - an.x, ah = an.w - an.y;
    float acx = an.x + 0.5f * aw, acy = an.y + 0.5f * ah;
    float awah = aw * ah;

    float best = -1.0f; int bg = 0;
    for (int g = 0; g < G; ++g) {
        float bx0 = sann[g*5+0], by0 = sann[g*5+1];
        float bx1 = sann[g*5+2], by1 = sann[g*5+3];
        float iw = fminf(an.z, bx1) - fmaxf(an.x, bx0);
        float ih = fminf(an.w, by1) - fmaxf(an.y, by0);
        iw = fmaxf(iw, 0.0f); ih = fmaxf(ih, 0.0f);
        float inter = iw * ih;
        float area  = (bx1 - bx0) * (by1 - by0);
        float ua    = fmaxf(awah + area - inter, 1e-8f);
        float iou   = inter / ua;
        if (iou > best) { best = iou; bg = g; }   // strict > keeps first max (jnp argmax)
    }

    bool  pos = best >= 0.5f;
    float lab = sann[bg*5+4];
    float tgt = pos ? lab : (best < 0.4f ? BG_CLASS : -1.0f);
    targets[(long)b * A + a] = tgt;

    if (pos) {
        float bx0 = sann[bg*5+0], by0 = sann[bg*5+1];
        float bx1 = sann[bg*5+2], by1 = sann[bg*5+3];
        float rgw = bx1 - bx0, rgh = by1 - by0;
        float gcx = bx0 + 0.5f * rgw, gcy = by0 + 0.5f * rgh;
        float gw = fmaxf(rgw, 1.0f), gh = fmaxf(rgh, 1.0f);
        float t0 = ((gcx - acx) / aw) * 10.0f;        // /0.1
        float t1 = ((gcy - acy) / ah) * 10.0f;
        float t2 = logf(gw / aw) * 5.0f;              // /0.2
        float t3 = logf(gh / ah) * 5.0f;
        const float* rg = regs + ((long)b * A + a) * 4;
        float tt[4] = {t0, t1, t2, t3};
        float s = 0.0f;
        #pragma unroll
        for (int k = 0; k < 4; ++k) {
            float d = fabsf(tt[k] - rg[k]);
            s += (d <= 1.0f / 9.0f) ? 4.5f * d * d : d - 0.5f / 9.0f;
        }
        atomicAdd(&regsum[b], s);
        atomicAdd(&nposv[b], 1.0f);
    }
}

// ---------------------------------------------------------------------------
// Kernel 2: class-prototype masked sums over support rows (streams 131 MB).
// 256 threads = 4 groups x 64 dims; group reads one 256B row coalesced.
// Block-level LDS reduction, then one global atomic per (class,dim) per block.
// ---------------------------------------------------------------------------
__global__ void fl_proto_sum_kernel(const float* __restrict__ emb,     // (B*A,64)
                                    const float* __restrict__ targets, // (B*A)
                                    float* __restrict__ protosum,      // 4*64
                                    float* __restrict__ protocnt,      // 4
                                    long ns) {
    __shared__ float ssum[4 * 64];
    __shared__ float scnt[4];
    int tid = threadIdx.x;
    if (tid < 4) scnt[tid] = 0.0f;
    ssum[tid] = 0.0f;
    __syncthreads();

    int  d    = tid & 63;
    int  grp  = tid >> 6;
    long gid  = (long)blockIdx.x * 4 + grp;
    long ngrp = (long)gridDim.x * 4;

    float acc0 = 0.f, acc1 = 0.f, acc2 = 0.f, acc3 = 0.f;
    float c0 = 0.f, c1 = 0.f, c2 = 0.f, c3 = 0.f;
    for (long r = gid; r < ns; r += ngrp) {
        float lab = targets[r];
        int k = (lab == 0.0f) ? 0 : (lab == 1.0f) ? 1 : (lab == 2.0f) ? 2
              : (lab == BG_CLASS) ? 3 : -1;
        if (k < 0) continue;                         // ignored rows (-1)
        float v = emb[r * 64 + d];
        if      (k == 0) { acc0 += v; c0 += 1.f; }
        else if (k == 1) { acc1 += v; c1 += 1.f; }
        else if (k == 2) { acc2 += v; c2 += 1.f; }
        else             { acc3 += v; c3 += 1.f; }
    }
    atomicAdd(&ssum[0 * 64 + d], acc0);
    atomicAdd(&ssum[1 * 64 + d], acc1);
    atomicAdd(&ssum[2 * 64 + d], acc2);
    atomicAdd(&ssum[3 * 64 + d], acc3);
    if (d == 0) {                                    // one counter per row-group
        atomicAdd(&scnt[0], c0); atomicAdd(&scnt[1], c1);
        atomicAdd(&scnt[2], c2); atomicAdd(&scnt[3], c3);
    }
    __syncthreads();
    atomicAdd(&protosum[tid], ssum[tid]);
    if (tid < 4) atomicAdd(&protocnt[tid], scnt[tid]);
}

// ---------------------------------------------------------------------------
// Kernel 3: finalize prototypes and their squared norms.
// ---------------------------------------------------------------------------
__global__ void fl_proto_fin_kernel(const float* __restrict__ protosum,
                                    const float* __restrict__ protocnt,
                                    float* __restrict__ proto,   // 4*64
                                    float* __restrict__ pn2) {   // 4
    __shared__ float sp[4 * 64];
    int tid = threadIdx.x;                 // 256
    int k = tid >> 6;
    float p = protosum[tid] / fmaxf(protocnt[k], 1.0f);
    proto[tid] = p;
    sp[tid] = p * p;
    __syncthreads();
    if (tid < 4) {
        float s = 0.0f;
        for (int i = 0; i < 64; ++i) s += sp[tid * 64 + i];
        pn2[tid] = s;
    }
}

// ---------------------------------------------------------------------------
// Kernel 4: query focal loss via V_WMMA_F32_16X16X4_F32.
// logits_k = 2*(q . p_k) - |p_k|^2  (the |q|^2 softmax term cancels).
// Each wave owns 16-query tiles: A = 16x4 query slice (ISA 32-bit A layout),
// B = 4x16 with proto_n in column n (n<4), chained over 16 K-steps.
// All control flow at the WMMA is wave-uniform => EXEC all ones (ISA req).
// ---------------------------------------------------------------------------
__global__ void fl_query_loss_kernel(const float* __restrict__ embq,     // (nq,64)
                                     const float* __restrict__ targetsq, // (nq)
                                     const float* __restrict__ proto,    // 4*64
                                     const float* __restrict__ pn2,      // 4
                                     float* __restrict__ clsacc,         // [sum,npq]
                                     long nq) {
    __shared__ float xT[8][64];            // per-wave 16x4 transpose buffer
    int tid  = threadIdx.x;
    int lane = tid & 31;
    int wv   = tid >> 5;                   // wave in block (blockDim=256 -> 8)
    int n    = lane & 15;                  // B/C/D column for this lane
    int k0   = (lane >> 4) << 1;           // K sub-offset: lanes 0-15 -> 0, 16-31 -> 2

    // Hoist B fragments (prototypes, constant across tiles): B[k][n] = proto_n[k]
    v2f bfr[16];
    #pragma unroll
    for (int s = 0; s < 16; ++s) {
        if (n < 4) {
            bfr[s].x = proto[n * 64 + 4 * s + k0];
            bfr[s].y = proto[n * 64 + 4 * s + k0 + 1];
        } else {
            bfr[s].x = 0.0f; bfr[s].y = 0.0f;
        }
    }
    float q0 = pn2[0], q1 = pn2[1], q2 = pn2[2], q3 = pn2[3];

    long nTiles = nq >> 4;                 // nq divisible by 16 here
    long wid = (long)blockIdx.x * (blockDim.x >> 5) + wv;
    long nw  = (long)gridDim.x * (blockDim.x >> 5);

    float clsSum = 0.0f, npqSum = 0.0f;
    for (long t = wid; t < nTiles; t += nw) {
        long qbase = t << 4;
        const float* rowp = embq + (qbase + (long)n) * 64 + k0;

        v8f c = {0.f, 0.f, 0.f, 0.f, 0.f, 0.f, 0.f, 0.f};
        #pragma unroll
        for (int s = 0; s < 16; ++s) {
            v2f a;
            a.x = rowp[4 * s];
            a.y = rowp[4 * s + 1];
            // D[m][n] += sum_k A[m][k]*B[k][n]   (16x16x4, exact f32)
            c = __builtin_amdgcn_wmma_f32_16x16x4_f32(
                    false, a, false, bfr[s], (short)0, c, false, false);
        }

        // Transpose the 16x4 useful block through wave-private LDS.
        if (n < 4) {
            int mhi = (lane >> 4) << 3;    // high lanes hold rows M=8..15
            #pragma unroll
            for (int j = 0; j < 8; ++j)
                xT[wv][(mhi + j) * 4 + n] = c[j];
        }
        __builtin_amdgcn_wave_barrier();
        asm volatile("s_wait_dscnt 0x0" ::: "memory");
        __builtin_amdgcn_wave_barrier();

        if (lane < 16) {
            int m = lane;
            float lab = targetsq[qbase + m];
            if (lab != -1.0f) {
                float l0 = 2.0f * xT[wv][m * 4 + 0] - q0;
                float l1 = 2.0f * xT[wv][m * 4 + 1] - q1;
                float l2 = 2.0f * xT[wv][m * 4 + 2] - q2;
                float l3 = 2.0f * xT[wv][m * 4 + 3] - q3;
                float mx = fmaxf(fmaxf(l0, l1), fmaxf(l2, l3));
                float e0 = expf(l0 - mx), e1 = expf(l1 - mx);
                float e2 = expf(l2 - mx), e3 = expf(l3 - mx);
                float ssum = e0 + e1 + e2 + e3;
                int  tix = (lab == 1.0f) ? 1 : (lab == 2.0f) ? 2
                         : (lab == BG_CLASS) ? 3 : 0;
                float lt   = (tix == 0) ? l0 : (tix == 1) ? l1 : (tix == 2) ? l2 : l3;
                float logp = (lt - mx) - logf(ssum);
                float p    = expf(lt - mx) / ssum;
                float om   = 1.0f - p;
                clsSum += -F_ALPHA * om * om * logp;
                if (lab != BG_CLASS) npqSum += 1.0f;
            }
        }
        __builtin_amdgcn_wave_barrier();   // xT WAR vs next tile (DS in-order per wave)
    }

    // Wave reduction (lanes 16-31 carry zeros).
    #pragma unroll
    for (int off = 16; off > 0; off >>= 1) {
        clsSum += __shfl_down(clsSum, off, 32);
        npqSum += __shfl_down(npqSum, off, 32);
    }
    if (lane == 0) {
        atomicAdd(&clsacc[0], clsSum);
        atomicAdd(&clsacc[1], npqSum);
    }
}

// ---------------------------------------------------------------------------
// Kernel 5: scalar finalize -> d_out = {cls_loss, reg_loss}.
// ---------------------------------------------------------------------------
__global__ void fl_final_kernel(const float* __restrict__ regsum,
                                const float* __restrict__ nposv,
                                const float* __restrict__ clsacc,
                                float* __restrict__ out, int B) {
    if (threadIdx.x == 0 && blockIdx.x == 0) {
        float r = 0.0f;
        for (int b = 0; b < B; ++b) {
            float np = nposv[b];
            r += (np > 0.0f) ? regsum[b] / fmaxf(4.0f * np, 1.0f) : 0.0f;
        }
        r /= (float)B;
        out[0] = clsacc[0] / fmaxf(clsacc[1], 1.0f);   // cls_loss
        out[1] = r;                                    // reg_loss (mean over images)
    }
}

// ---------------------------------------------------------------------------
extern "C" void kernel_launch(void* const* d_in, const int* in_sizes, int n_in,
                              void* d_out, int out_size, void* d_ws, size_t ws_size,
                              hipStream_t stream) {
    const float* cls     = (const float*)d_in[0];   // (B,A,64)
    const float* regs    = (const float*)d_in[1];   // (B,A,4)
    const float* anchors = (const float*)d_in[2];   // (1,A,4)
    const float* ann     = (const float*)d_in[3];   // (B,G,5)

    long A = (long)in_sizes[2] / 4;
    int  B = (int)((long)in_sizes[1] / (A * 4));
    int  G = (int)((long)in_sizes[3] / (B * 5));
    long BA = (long)B * A;
    long ns = 5L * A;            // support rows (NUM_SUPPORT_IMAGES = 5)
    long nq = BA - ns;           // query rows

    float* ws       = (float*)d_ws;
    float* targets  = ws;                       // BA floats
    float* accbase  = ws + BA;
    float* regsum   = accbase;                  // B
    float* nposv    = accbase + B;              // B
    float* protosum = accbase + 2 * B;          // 256
    float* protocnt = protosum + 256;           // 4
    float* proto    = protocnt + 4;             // 256
    float* pn2      = proto + 256;              // 4
    float* clsacc   = pn2 + 4;                  // 2
    int naccum = 2 * B + 256 + 4 + 256 + 4 + 2;

    fl_init_kernel<<<(naccum + 255) / 256, 256, 0, stream>>>(accbase, naccum);

    dim3 g1((unsigned)((A + 255) / 256), (unsigned)B);
    fl_assign_reg_kernel<<<g1, 256, 0, stream>>>(regs, anchors, ann, targets,
                                                 regsum, nposv, (int)A, G);

    fl_proto_sum_kernel<<<1024, 256, 0, stream>>>(cls, targets, protosum,
                                                  protocnt, ns);

    fl_proto_fin_kernel<<<1, 256, 0, stream>>>(protosum, protocnt, proto, pn2);

    fl_query_loss_kernel<<<480, 256, 0, stream>>>(cls + ns * 64, targets + ns,
                                                  proto, pn2, clsacc, nq);

    fl_final_kernel<<<1, 1, 0, stream>>>(regsum, nposv, clsacc, (float*)d_out, B);
}